// Warp_1786706395663
// MI455X (gfx1250) — compile-verified
//
#include <hip/hip_runtime.h>

// Bilinear backward warp, channels-last: out[b,h,w,c] = bilerp(img, (w,h)+flo[b,h,w])
// B=4, H=384, W=512, C=64, fp32. Memory-bound (AI < 1 FLOP/B) -> tuned for the
// MI455X memory system: coalesced b128 gathers, NT stores to preserve the
// 192MB L2 for the 201MB img working set, async global->LDS flow staging.

#define B_N 4
#define H_N 384
#define W_N 512
#define C_N 64
#define PIX_PER_BLOCK 64   // one block covers 64 consecutive pixels of ONE row
#define THREADS 256        // 8 wave32s; 16 lanes cover 64 channels as float4

typedef float v4f __attribute__((ext_vector_type(4)));

__global__ __launch_bounds__(THREADS) void warp_bilinear_kernel(
    const float* __restrict__ img,
    const float* __restrict__ flo,
    float* __restrict__ out)
{
    __shared__ float2 sflow[PIX_PER_BLOCK];   // 512 B of flow for this block

    const unsigned tid      = threadIdx.x;
    const unsigned blockPix = blockIdx.x * PIX_PER_BLOCK;

    // Uniform (SALU) decomposition: all 64 pixels share one image row.
    const unsigned row   = blockIdx.x >> 3;          // 512/64 = 8 blocks per row
    const unsigned wBase = (blockIdx.x & 7u) * PIX_PER_BLOCK;
    const unsigned b     = row / H_N;                // scalar div
    const unsigned h     = row - b * H_N;

    // --- CDNA5 async global->LDS: wave 0 stages 64 pixels x 8B of flow ------
    if (tid < 32u) {
        unsigned lds_off = (unsigned)(size_t)(&sflow[0]) + tid * 16u;
        unsigned long long gaddr =
            (unsigned long long)(size_t)flo +
            (unsigned long long)(blockPix + tid * 2u) * 8ull;
        asm volatile("global_load_async_to_lds_b128 %0, %1, off"
                     :: "v"(lds_off), "v"(gaddr) : "memory");
        asm volatile("s_wait_asynccnt 0" ::: "memory");
        // speculative prefetch of the next block's flow tile into GL2
        __builtin_prefetch(flo + (size_t)(blockPix + PIX_PER_BLOCK + tid * 2u) * 2, 0, 0);
    }
    __syncthreads();

    const unsigned sub    = tid & 15u;   // lane-in-pixel: 16 lanes x float4 = 64 ch
    const unsigned pLocal = tid >> 4;    // first pixel index within block
    const unsigned ch     = sub * 4u;

    const float*   imgB  = img + (size_t)b * ((size_t)H_N * W_N * C_N) + ch;
    float*         outB  = out + (size_t)blockPix * C_N + ch;

#pragma unroll
    for (int rep = 0; rep < 4; ++rep) {
        const unsigned pl = pLocal + (unsigned)rep * 16u;
        const unsigned w  = wBase + pl;

        const float2 f = sflow[pl];
        const float x = (float)w + f.x;
        const float y = (float)h + f.y;

        // trunc-toward-zero cast, matching jnp astype(int32)
        int x0 = (int)x, y0 = (int)y;
        int x1 = x0 + 1, y1 = y0 + 1;
        x0 = min(max(x0, 0), W_N - 1);
        x1 = min(max(x1, 0), W_N - 1);
        y0 = min(max(y0, 0), H_N - 1);
        y1 = min(max(y1, 0), H_N - 1);

        // weights from CLAMPED integer coords (matches reference)
        const float x0f = (float)x0, x1f = (float)x1;
        const float y0f = (float)y0, y1f = (float)y1;
        const float wa = (x1f - x) * (y1f - y);
        const float wb = (x1f - x) * (y - y0f);
        const float wc = (x - x0f) * (y1f - y);
        const float wd = (x - x0f) * (y - y0f);

        const float* r0 = imgB + (size_t)y0 * (W_N * C_N);
        const float* r1 = imgB + (size_t)y1 * (W_N * C_N);

        // four coalesced 256B-per-pixel b128 gathers (RT: L2-cached, img ~fits L2)
        const v4f Ia = *(const v4f*)(r0 + (size_t)x0 * C_N);
        const v4f Ic = *(const v4f*)(r0 + (size_t)x1 * C_N);
        const v4f Ib = *(const v4f*)(r1 + (size_t)x0 * C_N);
        const v4f Id = *(const v4f*)(r1 + (size_t)x1 * C_N);

        const v4f o = wa * Ia + wb * Ib + wc * Ic + wd * Id;  // contracts to v_fma

        // write-once stream: non-temporal store, keep L2 for img gathers
        __builtin_nontemporal_store(o, (v4f*)(outB + (size_t)pl * C_N));
    }
}

extern "C" void kernel_launch(void* const* d_in, const int* in_sizes, int n_in,
                              void* d_out, int out_size, void* d_ws, size_t ws_size,
                              hipStream_t stream)
{
    const float* img = (const float*)d_in[0];
    const float* flo = (const float*)d_in[1];
    float* out = (float*)d_out;

    const int totalPix = B_N * H_N * W_N;          // 786432
    dim3 grid(totalPix / PIX_PER_BLOCK);           // 12288 blocks, exact
    warp_bilinear_kernel<<<grid, THREADS, 0, stream>>>(img, flo, out);
}